// STEMProbeFunction_65429531787605
// MI455X (gfx1250) — compile-verified
//
#include <hip/hip_runtime.h>
#include <stdint.h>

typedef __attribute__((ext_vector_type(2))) float v2f;
typedef __attribute__((ext_vector_type(8))) float v8f;
typedef __attribute__((ext_vector_type(4))) unsigned int v4u;
typedef __attribute__((ext_vector_type(4))) int v4i;
typedef __attribute__((ext_vector_type(8))) int v8i;

#define HH 1024
#define WW 1024
#define BB 32
#define KS 15
#define HALO 7
#define TILE 16
#define WAVES 8
#define BLKW (WAVES * TILE)          // 128 output columns per block
#define SROWS (TILE + KS - 1)        // 30 staged rows
#define SCOLS (BLKW + 2 * HALO + 2)  // 144 staged cols (2 zero cols pad K to 32)
#define NCHUNK 8                     // 32 K-values / 4 per WMMA

#if defined(__has_builtin)
#if __has_builtin(__builtin_amdgcn_tensor_load_to_lds)
#define HAVE_TDM 1
#endif
#if __has_builtin(__builtin_amdgcn_s_wait_tensorcnt)
#define HAVE_WAIT_TENSOR 1
#endif
#endif

__global__ __launch_bounds__(256) void airy_conv_roll(
    const float* __restrict__ img, const float* __restrict__ kern,
    const int* __restrict__ shifts, float* __restrict__ out) {
  __shared__ float simg[SROWS * SCOLS];            // 17,280 B
  __shared__ float sB[KS * NCHUNK * 32 * 2];       // 30,720 B: Toeplitz kernel mats

  const int tid = threadIdx.x;
  const int lane = tid & 31;
  const int wave = tid >> 5;
  const int b = blockIdx.z;
  const int y0 = blockIdx.y * TILE;
  const int x0 = blockIdx.x * BLKW;
  const int gy0 = y0 - HALO;
  const int gx0 = x0 - HALO;
  const float* imgB = img + (size_t)b * ((size_t)HH * WW);

  const bool interior =
      (gy0 >= 0) && (gy0 + SROWS <= HH) && (gx0 >= 0) && (gx0 + SCOLS <= WW);

#if defined(HAVE_TDM)
  if (interior) {
    // --- Tensor Data Mover: one 2-D tile (144 x 30 f32) global -> LDS ---
    if (wave == 0) {
      uint32_t lds = (uint32_t)(uintptr_t)&simg[0];
      uint64_t ga = (uint64_t)(uintptr_t)(imgB + (size_t)gy0 * WW + gx0);
      v4u g0;
      g0.x = 1u;                                    // count=1 (valid user D#)
      g0.y = lds;                                   // lds_addr (bytes)
      g0.z = (uint32_t)ga;                          // global_addr[31:0]
      g0.w = (uint32_t)((ga >> 32) & 0x01FFFFFFu)   // global_addr[56:32]
             | (2u << 30);                          // type = 2 ("image")
      uint32_t td0 = (uint32_t)(WW - gx0);          // cols remaining from tile start
      uint32_t td1 = (uint32_t)(HH - gy0);          // rows remaining
      v8i g1;
      g1[0] = (int)(2u << 16);                      // data_size = 4 bytes
      g1[1] = (int)(td0 << 16);                     // tensor_dim0[15:0]
      g1[2] = (int)((td0 >> 16) | (td1 << 16));     // tensor_dim0[31:16] | tensor_dim1[15:0]
      g1[3] = (int)((td1 >> 16) | ((uint32_t)SCOLS << 16)); // tile_dim0 = 144
      g1[4] = (int)SROWS;                           // tile_dim1 = 30, tile_dim2 = 0
      g1[5] = (int)WW;                              // tensor_dim0_stride = W
      g1[6] = 0;
      g1[7] = 0;
      v4i gz = {0, 0, 0, 0};                        // 2-D tensor: groups 2/3 unused
#if __clang_major__ >= 23
      v8i gz8 = {0, 0, 0, 0, 0, 0, 0, 0};
      __builtin_amdgcn_tensor_load_to_lds(g0, g1, gz, gz, gz8, 0);
#else
      __builtin_amdgcn_tensor_load_to_lds(g0, g1, gz, gz, 0);
#endif
#if defined(HAVE_WAIT_TENSOR)
      __builtin_amdgcn_s_wait_tensorcnt(0);
#else
      asm volatile("s_wait_tensorcnt 0x0" ::: "memory");
#endif
    }
  } else
#endif
  {
    // Boundary (or no-TDM) path: predicated loads with zero halo fill.
    for (int e = tid; e < SROWS * SCOLS; e += 256) {
      int r = e / SCOLS;
      int c = e - r * SCOLS;
      int gy = gy0 + r;
      int gx = gx0 + c;
      float v = 0.0f;
      if (gy >= 0 && gy < HH && gx >= 0 && gx < WW)
        v = imgB[(size_t)gy * WW + gx];
      simg[e] = v;
    }
  }

  // --- Build banded Toeplitz kernel matrices in WMMA-B layout ---
  // sB[((ky*8+cb)*32 + lane)*2 + j] = B[K=2*(lane>>4)+j, N=lane&15] for chunk cb,
  // where B[k,n] = kern[ky, (cb*4+k) - n] if in [0,15) else 0.
  for (int idx = tid; idx < KS * NCHUNK * 32; idx += 256) {
    int l = idx & 31;
    int cb = (idx >> 5) & 7;
    int ky = idx >> 8;
    int n = l & 15;
    int h = l >> 4;
    int c0 = cb * 4 + 2 * h;
    int kx0 = c0 - n;
    int kx1 = c0 + 1 - n;
    sB[idx * 2 + 0] = (kx0 >= 0 && kx0 < KS) ? kern[ky * KS + kx0] : 0.0f;
    sB[idx * 2 + 1] = (kx1 >= 0 && kx1 < KS) ? kern[ky * KS + kx1] : 0.0f;
  }

  __syncthreads();

  // --- Per wave: one 16x16 output tile; conv = sum over ky of 8 WMMA f32 16x16x4 ---
  const int m = lane & 15;   // A-row / B-col / C-col index
  const int h2 = lane >> 4;  // lane half selects K pair
  const int xoff = wave * TILE;

  v8f acc = {};
  for (int ky = 0; ky < KS; ++ky) {
    const float* arow = &simg[(ky + m) * SCOLS + xoff + 2 * h2];
    const float* brow = &sB[(ky * NCHUNK * 32 + lane) * 2];
#pragma unroll
    for (int cb = 0; cb < NCHUNK; ++cb) {
      v2f a;
      a.x = arow[cb * 4];       // A[m, K=2*h2]
      a.y = arow[cb * 4 + 1];   // A[m, K=2*h2+1]
      v2f bm;
      bm.x = brow[cb * 64];     // B[K=2*h2,   n]
      bm.y = brow[cb * 64 + 1]; // B[K=2*h2+1, n]
      acc = __builtin_amdgcn_wmma_f32_16x16x4_f32(
          false, a, false, bm, (short)0, acc, false, false);
    }
  }

  // --- Fused scan-line roll on store: out[(x + s) & 1023] = conv[x] ---
  const int xt = x0 + xoff;
  const int ybase = y0 + 8 * h2;  // C/D layout: VGPR v -> row M = v + 8*(lane>>4)
  const int* srow = shifts + b * HH + ybase;
  float* outB = out + (size_t)b * ((size_t)HH * WW);
#pragma unroll
  for (int v = 0; v < 8; ++v) {
    int y = ybase + v;
    int s = srow[v];
    unsigned xc = (unsigned)(xt + m + s) & (WW - 1);
    outB[(size_t)y * WW + xc] = acc[v];
  }
}

extern "C" void kernel_launch(void* const* d_in, const int* in_sizes, int n_in,
                              void* d_out, int out_size, void* d_ws,
                              size_t ws_size, hipStream_t stream) {
  (void)in_sizes; (void)n_in; (void)out_size; (void)d_ws; (void)ws_size;
  const float* img = (const float*)d_in[0];
  const float* kern = (const float*)d_in[1];
  const int* shifts = (const int*)d_in[2];
  float* out = (float*)d_out;
  dim3 grid(WW / BLKW, HH / TILE, BB);
  airy_conv_roll<<<grid, dim3(256, 1, 1), 0, stream>>>(img, kern, shifts, out);
}